// MSCA_40475771797729
// MI455X (gfx1250) — compile-verified
//
#include <hip/hip_runtime.h>

#define CHN 64
#define HW 128
#define NPIX (HW*HW)      // 16384
#define TILE 32
#define RAW 54            // TILE + 2*(10+1)
#define RSTR 56
#define PH 52             // TILE + 2*10
#define PSTR 56

typedef __attribute__((ext_vector_type(2))) float v2f;
typedef __attribute__((ext_vector_type(8))) float v8f;

// ---------------------------------------------------------------------------
// Pass 1: per (batch, channel, 32x32 tile) compute
//   S = sum over {id, avg3, max3} of (dw5x5 + dw1x7*7x1 + dw1x11*11x1 + dw1x21*21x1)
// (all depthwise, biases included) and store S to workspace.
// ---------------------------------------------------------------------------
__global__ __launch_bounds__(256) void msca_pass1(
    const float* __restrict__ x,
    const float* __restrict__ w55,  const float* __restrict__ b55,
    const float* __restrict__ wh7,  const float* __restrict__ bh7,
    const float* __restrict__ wv7,  const float* __restrict__ bv7,
    const float* __restrict__ wh11, const float* __restrict__ bh11,
    const float* __restrict__ wv11, const float* __restrict__ bv11,
    const float* __restrict__ wh21, const float* __restrict__ bh21,
    const float* __restrict__ wv21, const float* __restrict__ bv21,
    float* __restrict__ S)
{
  __shared__ float raw[RAW*RSTR];     // input tile, halo 11, zero-padded
  __shared__ float pooled[PH*PSTR];   // pooled variant, halo 10 (0 outside image)
  __shared__ float tmp[PH*TILE];      // horizontal-conv scratch
  __shared__ float swA[25];
  __shared__ float swh[39];           // 7 | 11 | 21 at offsets 0,7,18
  __shared__ float swv[39];
  __shared__ float sb[7];             // b55, bh7, bv7, bh11, bv11, bh21, bv21

  const int tid = threadIdx.x;
  const int c   = blockIdx.y;
  const int tx  = (blockIdx.x & 3) * TILE;
  const int ty  = (blockIdx.x >> 2) * TILE;
  const float* xp = x + ((size_t)blockIdx.z * CHN + c) * NPIX;

  if (tid < 25)       { swA[tid] = w55[c*25 + tid]; }
  else if (tid < 32)  { int i = tid-25; swh[i]    = wh7 [c*7 +i]; swv[i]    = wv7 [c*7 +i]; }
  else if (tid < 43)  { int i = tid-32; swh[7+i]  = wh11[c*11+i]; swv[7+i]  = wv11[c*11+i]; }
  else if (tid < 64)  { int i = tid-43; swh[18+i] = wh21[c*21+i]; swv[18+i] = wv21[c*21+i]; }
  else if (tid == 64) {
    sb[0]=b55[c]; sb[1]=bh7[c]; sb[2]=bv7[c];
    sb[3]=bh11[c]; sb[4]=bv11[c]; sb[5]=bh21[c]; sb[6]=bv21[c];
  }

  for (int i = tid; i < RAW*RAW; i += 256) {
    int r = i / RAW, cc = i - r*RAW;
    int gh = ty + r - 11, gw = tx + cc - 11;
    float v = 0.f;
    if ((unsigned)gh < (unsigned)HW && (unsigned)gw < (unsigned)HW) v = xp[gh*HW + gw];
    raw[r*RSTR + cc] = v;
  }
  __syncthreads();

  const int row = tid >> 3;          // 0..31, each thread owns 4 consecutive cols
  const int c0  = (tid & 7) * 4;

  float stot[4] = {0.f, 0.f, 0.f, 0.f};

  for (int v = 0; v < 3; ++v) {
    // ---- build pooled variant (0 outside image => conv zero-padding) ----
    for (int i = tid; i < PH*PH; i += 256) {
      int r = i / PH, cc = i - r*PH;
      int gh = ty + r - 10, gw = tx + cc - 10;
      float pv = 0.f;
      if ((unsigned)gh < (unsigned)HW && (unsigned)gw < (unsigned)HW) {
        if (v == 0) {
          pv = raw[(r+1)*RSTR + (cc+1)];
        } else if (v == 1) {          // avg pool, count_include_pad (raw OOB = 0)
          float s = 0.f;
          #pragma unroll
          for (int dy = 0; dy < 3; ++dy)
            #pragma unroll
            for (int dx = 0; dx < 3; ++dx)
              s += raw[(r+dy)*RSTR + (cc+dx)];
          pv = s * (1.0f/9.0f);
        } else {                      // max pool, -inf padding => valid lanes only
          float m = -3.402823466e38f;
          #pragma unroll
          for (int dy = 0; dy < 3; ++dy)
            #pragma unroll
            for (int dx = 0; dx < 3; ++dx) {
              int nh = gh + dy - 1, nw = gw + dx - 1;
              if ((unsigned)nh < (unsigned)HW && (unsigned)nw < (unsigned)HW) {
                float t = raw[(r+dy)*RSTR + (cc+dx)];
                m = t > m ? t : m;
              }
            }
          pv = m;
        }
      }
      pooled[r*PSTR + cc] = pv;
    }
    __syncthreads();

    // ---- 5x5 depthwise ----
    float s0, s1, s2, s3;
    {
      float bb = sb[0];
      s0 = bb; s1 = bb; s2 = bb; s3 = bb;
      #pragma unroll
      for (int dy = 0; dy < 5; ++dy)
        #pragma unroll
        for (int dx = 0; dx < 5; ++dx) {
          float w = swA[dy*5 + dx];
          const float* pr = &pooled[(row+8+dy)*PSTR + (c0+8+dx)];
          s0 += w*pr[0]; s1 += w*pr[1]; s2 += w*pr[2]; s3 += w*pr[3];
        }
    }

    // ---- separable 1xK then Kx1 for K in {7, 11, 21} ----
    const int KS[3] = {7, 11, 21};
    const int KO[3] = {0, 7, 18};
    #pragma unroll
    for (int ki = 0; ki < 3; ++ki) {
      const int K = KS[ki], O = KO[ki], K2 = K >> 1;
      const float bh = sb[1 + 2*ki];
      for (int i = tid; i < PH*TILE; i += 256) {
        int r = i >> 5, cc = i & 31;
        float a = bh;
        const float* pr = &pooled[r*PSTR + cc + 10 - K2];
        #pragma unroll
        for (int d = 0; d < K; ++d) a += swh[O+d] * pr[d];
        tmp[i] = a;
      }
      __syncthreads();
      float a0 = sb[2 + 2*ki], a1 = a0, a2 = a0, a3 = a0;
      #pragma unroll
      for (int d = 0; d < K; ++d) {
        float w = swv[O+d];
        const float* tr = &tmp[(row + 10 - K2 + d)*TILE + c0];
        a0 += w*tr[0]; a1 += w*tr[1]; a2 += w*tr[2]; a3 += w*tr[3];
      }
      s0 += a0; s1 += a1; s2 += a2; s3 += a3;
      __syncthreads();   // tmp (and later pooled) safe to rewrite
    }
    stot[0] += s0; stot[1] += s1; stot[2] += s2; stot[3] += s3;
  }

  float* Sp = S + ((size_t)blockIdx.z * CHN + c) * NPIX + (ty+row)*HW + tx + c0;
  Sp[0] = stot[0]; Sp[1] = stot[1]; Sp[2] = stot[2]; Sp[3] = stot[3];
}

// ---------------------------------------------------------------------------
// Pass 2: out[b,co,p] = (sum_ci W[co,ci]*S[b,ci,p] + 3*b11[co]) * x[b,co,p]
// GEMM M=64, K=64, N=pixels via V_WMMA_F32_16X16X4_F32 (exact fp32).
// Wave covers 32 pixels x 64 c_out; block (8 waves) covers 256 pixels.
// ---------------------------------------------------------------------------
__global__ __launch_bounds__(256) void msca_pass2(
    const float* __restrict__ S,
    const float* __restrict__ x,
    const float* __restrict__ w11,
    const float* __restrict__ b11,
    float* __restrict__ out)
{
  __shared__ float Wl[CHN*CHN];
  __shared__ float bias3[CHN];
  const int tid = threadIdx.x;
  for (int i = tid; i < CHN*CHN; i += 256) Wl[i] = w11[i];
  if (tid < CHN) bias3[tid] = 3.0f * b11[tid];
  __syncthreads();

  const int wave = tid >> 5, lane = tid & 31;
  const int lo = lane & 15, hi = lane >> 4;

  const size_t gp  = (size_t)blockIdx.x * 256;       // pixel offset inside chunk
  const int bloc   = (int)(gp >> 14);                 // local batch
  const int p0     = (int)(gp & 16383) + wave * 32;   // wave's pixel base

  const float* Sb = S   + (size_t)bloc * ((size_t)CHN * NPIX);
  const float* xb = x   + (size_t)bloc * ((size_t)CHN * NPIX);
  float*       ob = out + (size_t)bloc * ((size_t)CHN * NPIX);

  v8f zero = {0.f,0.f,0.f,0.f,0.f,0.f,0.f,0.f};
  v8f acc[4][2];
  #pragma unroll
  for (int mt = 0; mt < 4; ++mt) { acc[mt][0] = zero; acc[mt][1] = zero; }

  const int pA = p0 + lo;
  const int pB = p0 + 16 + lo;

  #pragma unroll 4
  for (int kt = 0; kt < 16; ++kt) {
    // 32-bit 16x4 fragment: VGPR0 <-> K0 (lanes 0-15) / K2 (lanes 16-31); VGPR1 <-> K1/K3
    const int kk = kt*4 + hi*2;
    v2f bA, bB;
    bA[0] = Sb[(kk  )*NPIX + pA];
    bA[1] = Sb[(kk+1)*NPIX + pA];
    bB[0] = Sb[(kk  )*NPIX + pB];
    bB[1] = Sb[(kk+1)*NPIX + pB];
    #pragma unroll
    for (int mt = 0; mt < 4; ++mt) {
      v2f a = *(const v2f*)&Wl[(mt*16 + lo)*CHN + kk];   // 8B aligned (kk even)
      acc[mt][0] = __builtin_amdgcn_wmma_f32_16x16x4_f32(
          false, a, false, bA, (short)0, acc[mt][0], false, false);
      acc[mt][1] = __builtin_amdgcn_wmma_f32_16x16x4_f32(
          false, a, false, bB, (short)0, acc[mt][1], false, false);
    }
  }

  // C/D layout: VGPR r, lanes 0-15 -> M=r, lanes 16-31 -> M=8+r; N = lane&15
  #pragma unroll
  for (int mt = 0; mt < 4; ++mt)
    #pragma unroll
    for (int nt = 0; nt < 2; ++nt) {
      const int p = p0 + nt*16 + lo;
      #pragma unroll
      for (int r = 0; r < 8; ++r) {
        const int co  = mt*16 + hi*8 + r;
        const int idx = co*NPIX + p;
        ob[idx] = (acc[mt][nt][r] + bias3[co]) * xb[idx];
      }
    }
}

// ---------------------------------------------------------------------------
extern "C" void kernel_launch(void* const* d_in, const int* in_sizes, int n_in,
                              void* d_out, int out_size, void* d_ws, size_t ws_size,
                              hipStream_t stream)
{
  const float* x      = (const float*)d_in[0];
  const float* w55    = (const float*)d_in[1];
  const float* b55    = (const float*)d_in[2];
  const float* w17_0  = (const float*)d_in[3];
  const float* b17_0  = (const float*)d_in[4];
  const float* w17_1  = (const float*)d_in[5];
  const float* b17_1  = (const float*)d_in[6];
  const float* w111_0 = (const float*)d_in[7];
  const float* b111_0 = (const float*)d_in[8];
  const float* w111_1 = (const float*)d_in[9];
  const float* b111_1 = (const float*)d_in[10];
  const float* w211_0 = (const float*)d_in[11];
  const float* b211_0 = (const float*)d_in[12];
  const float* w211_1 = (const float*)d_in[13];
  const float* b211_1 = (const float*)d_in[14];
  const float* w11    = (const float*)d_in[15];
  const float* b11    = (const float*)d_in[16];
  float* out = (float*)d_out;
  float* S   = (float*)d_ws;

  const int B = 16;
  const size_t planeFloats   = (size_t)CHN * NPIX;                 // per-batch S
  const size_t perBatchBytes = planeFloats * sizeof(float);        // 4 MiB
  int chunk = (int)(ws_size / perBatchBytes);
  if (chunk < 1) chunk = 1;
  if (chunk > B) chunk = B;

  for (int b0 = 0; b0 < B; b0 += chunk) {
    const int nb = (B - b0 < chunk) ? (B - b0) : chunk;
    const float* xb   = x   + (size_t)b0 * planeFloats;
    float*       outb = out + (size_t)b0 * planeFloats;
    dim3 g1(16, CHN, nb);
    msca_pass1<<<g1, 256, 0, stream>>>(xb,
        w55, b55, w17_0, b17_0, w17_1, b17_1,
        w111_0, b111_0, w111_1, b111_1,
        w211_0, b211_0, w211_1, b211_1, S);
    msca_pass2<<<nb * 64, 256, 0, stream>>>(S, xb, w11, b11, outb);
  }
}